// Mask_27702539059334
// MI455X (gfx1250) — compile-verified
//
#include <hip/hip_runtime.h>
#include <stdint.h>

typedef __attribute__((ext_vector_type(4))) float v4f;

namespace {
constexpr int kTokens        = 196;                       // 14*14 tokens
constexpr int kTokDim        = 14;
constexpr int kMaskCount     = 20;                        // ceil(196*0.1)
constexpr int kMaskDim       = 56;                        // 14*4
constexpr int kBatch         = 256;
constexpr int kQuadsPerRow   = 56;                        // 224/4
constexpr int kQuadsPerBatch = 3 * 224 * kQuadsPerRow;    // 37632
constexpr long long kMaskedElems = 256LL * 3 * 224 * 224; // 38,535,168
constexpr int kBands         = (3 * 224) / 16;            // 42 16-row bands per batch
}

// ---------------------------------------------------------------------------
// Kernel A: per-batch top-k mask. One block per batch row.
// Stages the 196 noise floats into LDS with the CDNA5 async global->LDS path
// (tracked by ASYNCcnt), then computes each token's strict rank by counting.
// token masked  <=>  |{j : noise[j] > noise[t]}| < 20  <=>  noise[t] >= kth
// (exact, including tie semantics).
// ---------------------------------------------------------------------------
__global__ __launch_bounds__(224) void topk_mask_kernel(
    const float* __restrict__ noise,
    float* __restrict__ mask_out,        // [B,56,56] region of d_out
    uint32_t* __restrict__ rowbits_out)  // [B,16] words, 14 used (bit tc of word tr)
{
    __shared__ float    s_noise[kTokens];
    __shared__ uint32_t s_rowbits[kTokDim];

    const int b = blockIdx.x;
    const int t = threadIdx.x;

    if (t < kTokDim) s_rowbits[t] = 0u;

    if (t < kTokens) {
        // Async copy: one dword from srow+goff into LDS at lds_addr; per-lane, EXEC-masked.
        const float* srow = noise + b * kTokens;
        unsigned lds_addr = (unsigned)(uintptr_t)(&s_noise[t]); // low 32 bits of flat LDS addr = LDS offset
        unsigned goff     = (unsigned)(t * 4);
        asm volatile("global_load_async_to_lds_b32 %0, %1, %2"
                     :
                     : "v"(lds_addr), "v"(goff), "s"(srow)
                     : "memory");
    }
    asm volatile("s_wait_asynccnt 0" ::: "memory");
    __syncthreads();

    if (t < kTokens) {
        const float v = s_noise[t];
        int cnt = 0;
#pragma unroll 4
        for (int j = 0; j < kTokens; ++j) cnt += (s_noise[j] > v) ? 1 : 0;
        const uint32_t m  = (cnt < kMaskCount) ? 1u : 0u;
        const int tr = t / kTokDim;
        const int tc = t - tr * kTokDim;
        if (m) atomicOr(&s_rowbits[tr], 1u << tc);

        // Expand token -> 4x4 block of the 56x56 mask output (b128 stores).
        const float fm = (float)m;
        v4f fmv = {fm, fm, fm, fm};
        float* base = mask_out + (long long)b * (kMaskDim * kMaskDim)
                    + (4 * tr) * kMaskDim + 4 * tc;
#pragma unroll
        for (int rr = 0; rr < 4; ++rr)
            *(v4f*)(base + rr * kMaskDim) = fmv;
    }
    __syncthreads();
    if (t < kTokDim) rowbits_out[b * 16 + t] = s_rowbits[t];
}

// ---------------------------------------------------------------------------
// Kernel B: streaming select. Block = one 16-row band of one (batch, channel)
// image: 224 = 14*16 so a band never crosses a channel boundary and all 16
// rows share one token row => the mask word is block-uniform (scalar load).
// 224 threads (7 wave32) x 4 quads each; every instruction is a fully
// coalesced NT b128 (each wave covers a contiguous 512 B segment).
// 308 MB stream > 192 MB L2 and zero reuse => non-temporal hints both ways.
// ---------------------------------------------------------------------------
__global__ __launch_bounds__(224) void apply_mask_kernel(
    const float* __restrict__ x,
    const float* __restrict__ fill_ptr,
    const uint32_t* __restrict__ rowbits,
    float* __restrict__ out)
{
    const int b    = blockIdx.y;          // batch
    const int band = blockIdx.x;          // 16-row band within c*224+h space
    const int tid  = threadIdx.x;         // 4 rows x 56 quads

    const int row672 = band * 16;         // c*224 + h  (band start)
    const int h      = row672 % 224;      // h>>4 uniform over the whole band

    const float    fill = *fill_ptr;                       // scalar load
    const uint32_t bits = rowbits[b * 16 + (h >> 4)];      // block-uniform -> s_load

    const int  qw  = tid % 56;                             // quad column
    const bool msk = (bits >> (qw >> 2)) & 1u;

    int q = b * kQuadsPerBatch + row672 * kQuadsPerRow + tid;  // first quad index
    const v4f* __restrict__ xin  = (const v4f*)x;
    v4f* __restrict__       outq = (v4f*)out;

#pragma unroll
    for (int it = 0; it < 4; ++it) {
        v4f v = __builtin_nontemporal_load(xin + q);
        v4f o;
        o.x = msk ? fill : v.x;
        o.y = msk ? fill : v.y;
        o.z = msk ? fill : v.z;
        o.w = msk ? fill : v.w;
        __builtin_nontemporal_store(o, outq + q);
        q += 4 * kQuadsPerRow;            // advance 4 rows
    }
}

extern "C" void kernel_launch(void* const* d_in, const int* in_sizes, int n_in,
                              void* d_out, int out_size, void* d_ws, size_t ws_size,
                              hipStream_t stream) {
    (void)in_sizes; (void)n_in; (void)out_size; (void)ws_size;

    const float* x     = (const float*)d_in[0];  // [256,3,224,224] f32
    const float* noise = (const float*)d_in[1];  // [256,196] f32
    const float* fill  = (const float*)d_in[2];  // scalar f32
    float* out = (float*)d_out;                  // masked images | mask, concatenated
    uint32_t* rowbits = (uint32_t*)d_ws;         // 256*16 u32 = 16 KB scratch

    topk_mask_kernel<<<kBatch, 224, 0, stream>>>(noise, out + kMaskedElems, rowbits);

    dim3 grid(kBands, kBatch);                   // 42 x 256 blocks
    apply_mask_kernel<<<grid, 224, 0, stream>>>(x, fill, rowbits, out);
}